// GraphAutoEncoder_14989435863365
// MI455X (gfx1250) — compile-verified
//
#include <hip/hip_runtime.h>

// ---------------------------------------------------------------------------
// CDNA5 (gfx1250) GraphAutoEncoder: wave32 + V_WMMA_F32_16X16X4_F32 chains.
// Kernel 1: encoder MLP -> latent norm -> Gabriel graph -> 2-layer GCN -> pool
//           (one 16-row WMMA tile == 2 batches of 8 nodes per wave iteration)
// Kernel 2: decoder MLP as a dense WMMA GEMM over batches (16 batches/tile).
// ---------------------------------------------------------------------------

typedef float v2f __attribute__((ext_vector_type(2)));
typedef float v8f __attribute__((ext_vector_type(8)));

__device__ __forceinline__ v8f wmma4(v2f a, v2f b, v8f c) {
  // D(16x16,f32) = A(16x4,f32) * B(4x16,f32) + C
  return __builtin_amdgcn_wmma_f32_16x16x4_f32(false, a, false, b, (short)0, c,
                                               false, false);
}

__device__ __forceinline__ v8f splat8(float x) {
  v8f c = {x, x, x, x, x, x, x, x};
  return c;
}

__device__ __forceinline__ void lds_fence() {
  asm volatile("s_wait_dscnt 0" ::: "memory");
}

// Pack a row-major weight W[Kreal][Nreal] into WMMA B-operand layout in LDS:
// packed index p = (s*NT + j)*32 + lane, two floats per p:
//   B[k0][16j + (lane&15)], B[k0+1][...] with k0 = 4s + (lane>=16 ? 2 : 0).
// Out-of-range K/N padded with zeros (handles K=3->4, N=2->16 etc.).
__device__ __forceinline__ void pack_B(float* dst, const float* W, int Kreal,
                                       int Nreal, int S, int NT, int tid,
                                       int nthreads) {
  int total = S * NT * 32;
  for (int p = tid; p < total; p += nthreads) {
    int s = p / (NT * 32);
    int rem = p - s * NT * 32;
    int j = rem >> 5;
    int l = rem & 31;
    int col = j * 16 + (l & 15);
    int k0 = 4 * s + ((l >= 16) ? 2 : 0);
    dst[2 * p + 0] = (k0 < Kreal && col < Nreal) ? W[k0 * Nreal + col] : 0.0f;
    dst[2 * p + 1] =
        (k0 + 1 < Kreal && col < Nreal) ? W[(k0 + 1) * Nreal + col] : 0.0f;
  }
}

__device__ __forceinline__ void pad_copy(float* dst, const float* src,
                                         int nreal, int npad, int tid,
                                         int nthreads) {
  for (int i = tid; i < npad; i += nthreads)
    dst[i] = (i < nreal) ? src[i] : 0.0f;
}

// ----------------------------- Kernel 1 ------------------------------------
// LDS layout (floats):
//  We1B 256 | We2B 1024 | Wg1B 128 | Wg2B 1024 | be1 64 | be2p 16 | bg1 32 |
//  bg2 32  -> 2576 shared,  then per-wave: stage[16*64]=1024, ahat[256],
//  lats[32], dinv[16] -> 1344 (padded) * 8 waves.
#define K1_SHW 2576
#define K1_PW 1344
#define K1_TOT (K1_SHW + 8 * K1_PW)

__global__ __launch_bounds__(256) void gae_enc(
    const float* __restrict__ x, const float* __restrict__ noise,
    const float* __restrict__ We1, const float* __restrict__ be1,
    const float* __restrict__ We2, const float* __restrict__ be2,
    const float* __restrict__ Wg1, const float* __restrict__ bg1,
    const float* __restrict__ Wg2, const float* __restrict__ bg2,
    float* __restrict__ out_lat, float* __restrict__ out_A,
    float* __restrict__ pooled_ws, int Btot) {
  __shared__ __attribute__((aligned(16))) float sm[K1_TOT];
  float* We1B = sm;            // S=1, NT=4
  float* We2B = We1B + 256;    // S=16, NT=1 (N padded 2->16)
  float* Wg1B = We2B + 1024;   // S=1, NT=2
  float* Wg2B = Wg1B + 128;    // S=8, NT=2
  float* be1p = Wg2B + 1024;   // 64
  float* be2p = be1p + 64;     // 16 (padded)
  float* bg1p = be2p + 16;     // 32
  float* bg2p = bg1p + 32;     // 32

  int tid = threadIdx.x;
  int nth = blockDim.x;
  pack_B(We1B, We1, 3, 64, 1, 4, tid, nth);
  pack_B(We2B, We2, 64, 2, 16, 1, tid, nth);
  pack_B(Wg1B, Wg1, 2, 32, 1, 2, tid, nth);
  pack_B(Wg2B, Wg2, 32, 32, 8, 2, tid, nth);
  pad_copy(be1p, be1, 64, 64, tid, nth);
  pad_copy(be2p, be2, 2, 16, tid, nth);
  pad_copy(bg1p, bg1, 32, 32, tid, nth);
  pad_copy(bg2p, bg2, 32, 32, tid, nth);
  __syncthreads();

  int wave = tid >> 5;
  float* wsm = sm + K1_SHW + wave * K1_PW;
  float* stage = wsm;          // [16][64] max staging
  float* ahat = wsm + 1024;    // [16][16] block-diag normalized adjacency
  float* lats = wsm + 1280;    // [16][2] latent points
  float* dinv = wsm + 1312;    // [16]

  int lane = tid & 31;
  int m = lane & 15;           // row in tile / col in B,C layouts
  bool hi = lane >= 16;
  int n = m;                   // column index for B/C layouts
  int koff = hi ? 2 : 0;

  int wid = ((int)blockIdx.x * (int)blockDim.x + tid) >> 5;
  int nwaves = ((int)gridDim.x * (int)blockDim.x) >> 5;
  int tiles = Btot >> 1;       // 2 batches per 16-row tile

  for (int t = wid; t < tiles; t += nwaves) {
    // ---------------- encoder layer 1: feats(16x4) @ We1 -> 16x64 ----------
    float xv = x[t * 16 + m];
    v2f af;
    af.x = hi ? (float)(m & 7) : 0.0f;  // K0 = 0   | K2 = node index
    af.y = hi ? 0.0f : xv;              // K1 = x   | K3 = 0 (pad)
    v8f h[4];
    for (int j = 0; j < 4; ++j) {
      v2f b = *(const v2f*)(We1B + (j * 32 + lane) * 2);
      v8f c = splat8(be1p[j * 16 + n]);
      c = wmma4(af, b, c);
      for (int r = 0; r < 8; ++r) c[r] = fmaxf(c[r], 0.0f);
      h[j] = c;
    }
    // C-layout -> row-major LDS stage[16][64]
    {
      int row0 = hi ? 8 : 0;
      for (int j = 0; j < 4; ++j)
        for (int r = 0; r < 8; ++r)
          stage[(row0 + r) * 64 + j * 16 + n] = h[j][r];
    }
    lds_fence();
    // ---------------- encoder layer 2: h(16x64) @ We2 -> lat (cols 0,1) ----
    {
      v8f c = splat8(be2p[n]);
      for (int s = 0; s < 16; ++s) {
        v2f aa = *(const v2f*)(stage + m * 64 + 4 * s + koff);
        v2f bb = *(const v2f*)(We2B + (s * 32 + lane) * 2);
        c = wmma4(aa, bb, c);
      }
      if (n < 2) {
        int row0 = hi ? 8 : 0;
        for (int r = 0; r < 8; ++r) lats[(row0 + r) * 2 + n] = c[r];
      }
    }
    lds_fence();
    // ---------------- center / scale by 3/std / add noise ------------------
    {
      int row = m;            // 0..15 (batch = row>>3)
      int d = hi ? 1 : 0;
      float v = lats[row * 2 + d];
      float s1 = v;
      s1 += __shfl_xor(s1, 1, 32);
      s1 += __shfl_xor(s1, 2, 32);
      s1 += __shfl_xor(s1, 4, 32);
      float cdev = v - s1 * 0.125f;
      float s2 = cdev * cdev;
      s2 += __shfl_xor(s2, 1, 32);
      s2 += __shfl_xor(s2, 2, 32);
      s2 += __shfl_xor(s2, 4, 32);
      float stdv = sqrtf(s2 * (1.0f / 7.0f)) + 1e-8f;
      float latv =
          cdev * (3.0f / stdv) + noise[(t * 16 + row) * 2 + d] * 0.05f;
      out_lat[(t * 16 + row) * 2 + d] = latv;
      lats[row * 2 + d] = latv;
    }
    lds_fence();
    // ---------------- Gabriel graph + GCN normalization --------------------
    for (int q = 0; q < 8; ++q) ahat[q * 32 + lane] = 0.0f;  // zero 16x16
    lds_fence();
    for (int it = 0; it < 4; ++it) {
      int p = it * 32 + lane;  // 2 batches * 64 (i,j) pairs
      int bb = p >> 6;
      int i = (p >> 3) & 7;
      int j = p & 7;
      int ri = bb * 8 + i, rj = bb * 8 + j;
      float pix = lats[ri * 2 + 0], piy = lats[ri * 2 + 1];
      float pjx = lats[rj * 2 + 0], pjy = lats[rj * 2 + 1];
      float mx = (pix + pjx) * 0.5f, my = (piy + pjy) * 0.5f;
      float dx = pix - mx, dy = piy - my;
      float r2 = dx * dx + dy * dy;
      bool ok = true;
      for (int k = 0; k < 8; ++k) {
        float kx = lats[(bb * 8 + k) * 2 + 0];
        float ky = lats[(bb * 8 + k) * 2 + 1];
        float ex = kx - mx, ey = ky - my;
        bool excl = (k == i) || (k == j);
        ok = ok && (excl || (ex * ex + ey * ey >= r2));
      }
      float Aij = (i == j) ? 0.0f : (ok ? 1.0f : 0.0f);
      out_A[(t * 2 + bb) * 64 + i * 8 + j] = Aij;
      ahat[ri * 16 + rj] = Aij + ((i == j) ? 1.0f : 0.0f);  // A + I
    }
    lds_fence();
    if (lane < 16) {
      float dsum = 0.0f;
      for (int j2 = 0; j2 < 16; ++j2) dsum += ahat[lane * 16 + j2];
      dinv[lane] = 1.0f / sqrtf(dsum);
    }
    lds_fence();
    for (int q = 0; q < 8; ++q) {
      int e = q * 32 + lane;
      ahat[e] = ahat[e] * dinv[e >> 4] * dinv[e & 15];
    }
    lds_fence();
    // ---------------- GCN layer 1 ------------------------------------------
    v2f al;  // lat as A-operand, K padded 2->4
    al.x = hi ? 0.0f : lats[m * 2 + 0];
    al.y = hi ? 0.0f : lats[m * 2 + 1];
    v8f xw[2];
    for (int j = 0; j < 2; ++j) {
      v2f bb = *(const v2f*)(Wg1B + (j * 32 + lane) * 2);
      xw[j] = wmma4(al, bb, splat8(0.0f));
    }
    {
      int row0 = hi ? 8 : 0;
      for (int j = 0; j < 2; ++j)
        for (int r = 0; r < 8; ++r)
          stage[(row0 + r) * 32 + j * 16 + n] = xw[j][r];
    }
    lds_fence();
    v8f g1[2];  // relu(A_hat @ xw + bg1), A_hat as WMMA A-operand
    for (int j = 0; j < 2; ++j) {
      v8f c = splat8(bg1p[j * 16 + n]);
      for (int s = 0; s < 4; ++s) {
        v2f aa = *(const v2f*)(ahat + m * 16 + 4 * s + koff);
        int k0 = 4 * s + koff;
        v2f bb;
        bb.x = stage[k0 * 32 + j * 16 + n];
        bb.y = stage[(k0 + 1) * 32 + j * 16 + n];
        c = wmma4(aa, bb, c);
      }
      for (int r = 0; r < 8; ++r) c[r] = fmaxf(c[r], 0.0f);
      g1[j] = c;
    }
    lds_fence();
    // ---------------- GCN layer 2 ------------------------------------------
    {
      int row0 = hi ? 8 : 0;
      for (int j = 0; j < 2; ++j)
        for (int r = 0; r < 8; ++r)
          stage[(row0 + r) * 32 + j * 16 + n] = g1[j][r];
    }
    lds_fence();
    v8f gw[2];  // g1 @ Wg2
    for (int j = 0; j < 2; ++j) {
      v8f c = splat8(0.0f);
      for (int s = 0; s < 8; ++s) {
        v2f aa = *(const v2f*)(stage + m * 32 + 4 * s + koff);
        v2f bb = *(const v2f*)(Wg2B + ((s * 2 + j) * 32 + lane) * 2);
        c = wmma4(aa, bb, c);
      }
      gw[j] = c;
    }
    lds_fence();
    {
      int row0 = hi ? 8 : 0;
      for (int j = 0; j < 2; ++j)
        for (int r = 0; r < 8; ++r)
          stage[(row0 + r) * 32 + j * 16 + n] = gw[j][r];
    }
    lds_fence();
    // g2 = A_hat @ gw + bg2 ; pooled = mean over 8 nodes (row sum of C tile)
    for (int j = 0; j < 2; ++j) {
      v8f c = splat8(bg2p[j * 16 + n]);
      for (int s = 0; s < 4; ++s) {
        v2f aa = *(const v2f*)(ahat + m * 16 + 4 * s + koff);
        int k0 = 4 * s + koff;
        v2f bb;
        bb.x = stage[k0 * 32 + j * 16 + n];
        bb.y = stage[(k0 + 1) * 32 + j * 16 + n];
        c = wmma4(aa, bb, c);
      }
      float ps = 0.0f;
      for (int r = 0; r < 8; ++r) ps += c[r];
      ps *= 0.125f;
      pooled_ws[(t * 2 + (hi ? 1 : 0)) * 32 + j * 16 + n] = ps;
    }
    lds_fence();
  }
}

// ----------------------------- Kernel 2 ------------------------------------
#define K2_SHW 3152
#define K2_PW 1040
#define K2_TOT (K2_SHW + 8 * K2_PW)

__global__ __launch_bounds__(256) void gae_dec(
    const float* __restrict__ pooled_ws, const float* __restrict__ Wd1,
    const float* __restrict__ bd1, const float* __restrict__ Wd2,
    const float* __restrict__ bd2, float* __restrict__ out_recon, int Btot) {
  __shared__ __attribute__((aligned(16))) float sm[K2_TOT];
  float* Wd1B = sm;            // S=8, NT=4 -> 2048
  float* Wd2B = Wd1B + 2048;   // S=16, NT=1 (N padded 3->16) -> 1024
  float* bd1p = Wd2B + 1024;   // 64
  float* bd2p = bd1p + 64;     // 16

  int tid = threadIdx.x;
  int nth = blockDim.x;
  pack_B(Wd1B, Wd1, 32, 64, 8, 4, tid, nth);
  pack_B(Wd2B, Wd2, 64, 3, 16, 1, tid, nth);
  pad_copy(bd1p, bd1, 64, 64, tid, nth);
  pad_copy(bd2p, bd2, 3, 16, tid, nth);
  __syncthreads();

  int wave = tid >> 5;
  float* stage = sm + K2_SHW + wave * K2_PW;  // [16][64]
  float* small = stage + 1024;                // [16] dec[:,1]

  int lane = tid & 31;
  int m = lane & 15;
  bool hi = lane >= 16;
  int n = m;
  int koff = hi ? 2 : 0;

  int wid = ((int)blockIdx.x * (int)blockDim.x + tid) >> 5;
  int nwaves = ((int)gridDim.x * (int)blockDim.x) >> 5;
  int tiles = Btot >> 4;  // 16 batches per tile

  for (int t = wid; t < tiles; t += nwaves) {
    // layer 1: h = relu(pooled @ Wd1 + bd1), A rows cached in registers
    v2f arow[8];
    for (int s = 0; s < 8; ++s)
      arow[s] = *(const v2f*)(pooled_ws + (t * 16 + m) * 32 + 4 * s + koff);
    v8f h[4];
    for (int j = 0; j < 4; ++j) {
      v8f c = splat8(bd1p[j * 16 + n]);
      for (int s = 0; s < 8; ++s) {
        v2f bb = *(const v2f*)(Wd1B + ((s * 4 + j) * 32 + lane) * 2);
        c = wmma4(arow[s], bb, c);
      }
      for (int r = 0; r < 8; ++r) c[r] = fmaxf(c[r], 0.0f);
      h[j] = c;
    }
    {
      int row0 = hi ? 8 : 0;
      for (int j = 0; j < 4; ++j)
        for (int r = 0; r < 8; ++r)
          stage[(row0 + r) * 64 + j * 16 + n] = h[j][r];
    }
    lds_fence();
    // layer 2: dec = h @ Wd2 + bd2 ; only column 1 is the reconstruction
    {
      v8f c = splat8(bd2p[n]);
      for (int s = 0; s < 16; ++s) {
        v2f aa = *(const v2f*)(stage + m * 64 + 4 * s + koff);
        v2f bb = *(const v2f*)(Wd2B + (s * 32 + lane) * 2);
        c = wmma4(aa, bb, c);
      }
      if (n == 1) {
        int row0 = hi ? 8 : 0;
        for (int r = 0; r < 8; ++r) small[row0 + r] = c[r];
      }
    }
    lds_fence();
    // broadcast dec[:,1] to all 8 nodes: 128 consecutive floats, float4 each
    {
      float v = small[lane >> 1];
      float4 o = {v, v, v, v};
      *(float4*)(out_recon + t * 128 + lane * 4) = o;
    }
    lds_fence();
  }
}

// ----------------------------- launcher ------------------------------------
extern "C" void kernel_launch(void* const* d_in, const int* in_sizes, int n_in,
                              void* d_out, int out_size, void* d_ws,
                              size_t ws_size, hipStream_t stream) {
  const float* x = (const float*)d_in[0];
  const float* noise = (const float*)d_in[1];
  const float* We1 = (const float*)d_in[2];
  const float* be1 = (const float*)d_in[3];
  const float* We2 = (const float*)d_in[4];
  const float* be2 = (const float*)d_in[5];
  const float* Wg1 = (const float*)d_in[6];
  const float* bg1 = (const float*)d_in[7];
  const float* Wg2 = (const float*)d_in[8];
  const float* bg2 = (const float*)d_in[9];
  const float* Wd1 = (const float*)d_in[10];
  const float* bd1 = (const float*)d_in[11];
  const float* Wd2 = (const float*)d_in[12];
  const float* bd2 = (const float*)d_in[13];

  int Btot = in_sizes[0] / 8;                       // 131072
  float* out = (float*)d_out;
  float* out_recon = out;                           // [B,8]
  float* out_lat = out + (size_t)Btot * 8;          // [B,8,2]
  float* out_A = out_lat + (size_t)Btot * 16;       // [B,8,8]
  float* pooled = (float*)d_ws;                     // [B,32] scratch

  dim3 blk(256);
  int tiles1 = Btot >> 1;
  int grid1 = 2048;
  if (grid1 * 8 > tiles1) grid1 = (tiles1 + 7) / 8;
  gae_enc<<<grid1, blk, 0, stream>>>(x, noise, We1, be1, We2, be2, Wg1, bg1,
                                     Wg2, bg2, out_lat, out_A, pooled, Btot);

  int tiles2 = Btot >> 4;
  int grid2 = 1024;
  if (grid2 * 8 > tiles2) grid2 = (tiles2 + 7) / 8;
  gae_dec<<<grid2, blk, 0, stream>>>(pooled, Wd1, bd1, Wd2, bd2, out_recon,
                                     Btot);
}